// NGCFLayer_72928544686737
// MI455X (gfx1250) — compile-verified
//
#include <hip/hip_runtime.h>

typedef __attribute__((ext_vector_type(2))) float v2f;
typedef __attribute__((ext_vector_type(8))) float v8f;

#define DD 128
// k-paired W^T layout: lWp[k/2][n][2]; pair-row pitch in floats.
// 288 % 64 == 32 dwords -> adjacent pair-rows map to opposite LDS bank halves,
// so lanes 0-15 (pair row kp) and lanes 16-31 (pair row kp+1) are conflict-free.
#define KPITCH 288

// ---------------------------------------------------------------------------
// lin1: Y = X @ W^T + b  (fp32 WMMA 16x16x4). Writes w1 buffer AND h accumulator
// (h starts as the self-loop term w1 before edge messages are atomically added).
// One wave handles a 16-row tile of X; W^T (k-paired) lives in LDS.
// ---------------------------------------------------------------------------
__global__ __launch_bounds__(256) void ngcf_lin1_kernel(
    const float* __restrict__ X, const float* __restrict__ W,
    const float* __restrict__ bias, float* __restrict__ w1,
    float* __restrict__ hacc, int nrows)
{
  __shared__ float lWp[64 * KPITCH];             // lWp[kp*KPITCH + n*2 + j] = W[n][2kp+j]
  const int tid = threadIdx.x;
  for (int idx = tid; idx < DD * DD; idx += 256) {
    const int n = idx >> 7, k = idx & 127;       // W row-major: W[n][k]
    lWp[(k >> 1) * KPITCH + n * 2 + (k & 1)] = W[idx];
  }
  __syncthreads();

  const int wave = tid >> 5;
  const int lane = tid & 31;
  const int half = lane >> 4;                    // 0: lanes 0-15, 1: lanes 16-31
  const int ln   = lane & 15;
  const int row_base = (blockIdx.x * 8 + wave) * 16;
  if (row_base >= nrows) return;                 // wave-uniform: EXEC stays all-1s

  float bv[8];
#pragma unroll
  for (int nt = 0; nt < 8; ++nt) bv[nt] = bias[nt * 16 + ln];

  v8f zero = {};
  v8f acc[8];
#pragma unroll
  for (int nt = 0; nt < 8; ++nt) acc[nt] = zero;

  const float* xrow = X + (size_t)(row_base + ln) * DD;

  for (int k0 = 0; k0 < DD; k0 += 4) {
    // lanes 0-15 carry K=k0,k0+1 ; lanes 16-31 carry K=k0+2,k0+3
    const v2f a = *(const v2f*)(xrow + k0 + 2 * half);
    const float* brow = &lWp[((k0 >> 1) + half) * KPITCH];
#pragma unroll
    for (int nt = 0; nt < 8; ++nt) {
      const v2f b = *(const v2f*)(brow + (nt * 16 + ln) * 2);
      acc[nt] = __builtin_amdgcn_wmma_f32_16x16x4_f32(
          false, a, false, b, (short)0, acc[nt], false, false);
    }
  }

  // D layout: VGPR r -> row r (lanes 0-15) / row r+8 (lanes 16-31), col = ln in tile
#pragma unroll
  for (int r = 0; r < 8; ++r) {
    const size_t row = (size_t)(row_base + r + 8 * half);
    float* w1row = w1   + row * DD;
    float* harow = hacc + row * DD;
#pragma unroll
    for (int nt = 0; nt < 8; ++nt) {
      const int col = nt * 16 + ln;
      const float v = acc[nt][r] + bv[nt];
      w1row[col] = v;
      harow[col] = v;
    }
  }
}

// ---------------------------------------------------------------------------
// edge: per 16-edge tile, A = fu*fi (built on the fly), L2P = A @ W2^T (WMMA),
// then msg_ui = norm*(w1u[u]+b2+L2P) -> atomicAdd h_item[i]
//      msg_iu = norm*(w1i[i]+b2+L2P) -> atomicAdd h_user[u]
// ---------------------------------------------------------------------------
__global__ __launch_bounds__(256) void ngcf_edge_kernel(
    const float* __restrict__ feat_user, const float* __restrict__ feat_item,
    const float* __restrict__ edge_norm, const float* __restrict__ W2,
    const float* __restrict__ b2, const int* __restrict__ u_idx,
    const int* __restrict__ i_idx, const float* __restrict__ w1u,
    const float* __restrict__ w1i, float* __restrict__ h_user,
    float* __restrict__ h_item, int nedges)
{
  __shared__ float lWp[64 * KPITCH];             // W2^T, k-paired
  const int tid = threadIdx.x;
  for (int idx = tid; idx < DD * DD; idx += 256) {
    const int n = idx >> 7, k = idx & 127;
    lWp[(k >> 1) * KPITCH + n * 2 + (k & 1)] = W2[idx];
  }
  __syncthreads();

  const int wave = tid >> 5;
  const int lane = tid & 31;
  const int half = lane >> 4;
  const int ln   = lane & 15;
  const int ebase = (blockIdx.x * 8 + wave) * 16;
  if (ebase >= nedges) return;

  // lanes r and r+16 both hold metadata for edge (ebase + r)
  const int   e   = ebase + ln;
  const int   u   = u_idx[e];
  const int   it  = i_idx[e];
  const float nrm = edge_norm[e];

  float bv[8];
#pragma unroll
  for (int nt = 0; nt < 8; ++nt) bv[nt] = b2[nt * 16 + ln];

  const float* fu = feat_user + (size_t)u  * DD;
  const float* fi = feat_item + (size_t)it * DD;

  v8f zero = {};
  v8f acc[8];
#pragma unroll
  for (int nt = 0; nt < 8; ++nt) acc[nt] = zero;

  for (int k0 = 0; k0 < DD; k0 += 4) {
    const int k = k0 + 2 * half;
    const v2f au = *(const v2f*)(fu + k);
    const v2f ai = *(const v2f*)(fi + k);
    v2f a;
    a.x = au.x * ai.x;
    a.y = au.y * ai.y;
    const float* brow = &lWp[((k0 >> 1) + half) * KPITCH];
#pragma unroll
    for (int nt = 0; nt < 8; ++nt) {
      const v2f b = *(const v2f*)(brow + (nt * 16 + ln) * 2);
      acc[nt] = __builtin_amdgcn_wmma_f32_16x16x4_f32(
          false, a, false, b, (short)0, acc[nt], false, false);
    }
  }

  // scatter both message directions; broadcast per-edge metadata once per row
#pragma unroll
  for (int r = 0; r < 8; ++r) {
    const int row = r + 8 * half;                // edge index within tile
    const float nr = __shfl(nrm, row, 32);
    const int   ur = __shfl(u,   row, 32);
    const int   ir = __shfl(it,  row, 32);
    const float* w1urow = w1u + (size_t)ur * DD;
    const float* w1irow = w1i + (size_t)ir * DD;
    float* hirow = h_item + (size_t)ir * DD;
    float* hurow = h_user + (size_t)ur * DD;
#pragma unroll
    for (int nt = 0; nt < 8; ++nt) {
      const int col = nt * 16 + ln;
      const float l2p = acc[nt][r] + bv[nt];     // lin2(prod) incl. bias
      atomicAdd(hirow + col, nr * (w1urow[col] + l2p));  // user -> item msg
      atomicAdd(hurow + col, nr * (w1irow[col] + l2p));  // item -> user msg
    }
  }
}

// ---------------------------------------------------------------------------
// finish: LeakyReLU(0.2) then L2 row-normalize; one wave per row, shfl reduce.
// Output rows: [0, n_users) from h_user, then items.
// ---------------------------------------------------------------------------
__global__ __launch_bounds__(256) void ngcf_finish_kernel(
    const float* __restrict__ h_user, const float* __restrict__ h_item,
    float* __restrict__ out, int total_rows, int n_users)
{
  const int wave = threadIdx.x >> 5;
  const int lane = threadIdx.x & 31;
  const int row  = blockIdx.x * 8 + wave;
  if (row >= total_rows) return;

  const float* src = (row < n_users)
      ? (h_user + (size_t)row * DD)
      : (h_item + (size_t)(row - n_users) * DD);

  float4 v = reinterpret_cast<const float4*>(src)[lane];  // 32 lanes * 4 = 128
  v.x = v.x > 0.f ? v.x : 0.2f * v.x;
  v.y = v.y > 0.f ? v.y : 0.2f * v.y;
  v.z = v.z > 0.f ? v.z : 0.2f * v.z;
  v.w = v.w > 0.f ? v.w : 0.2f * v.w;

  float ss = v.x * v.x + v.y * v.y + v.z * v.z + v.w * v.w;
#pragma unroll
  for (int off = 16; off > 0; off >>= 1) ss += __shfl_xor(ss, off, 32);

  const float inv = 1.0f / fmaxf(sqrtf(ss), 1e-12f);
  v.x *= inv; v.y *= inv; v.z *= inv; v.w *= inv;
  reinterpret_cast<float4*>(out + (size_t)row * DD)[lane] = v;
}

// ---------------------------------------------------------------------------
extern "C" void kernel_launch(void* const* d_in, const int* in_sizes, int n_in,
                              void* d_out, int out_size, void* d_ws, size_t ws_size,
                              hipStream_t stream) {
  const float* feat_user = (const float*)d_in[0];
  const float* feat_item = (const float*)d_in[1];
  const float* edge_norm = (const float*)d_in[2];
  const float* W1_w      = (const float*)d_in[3];
  const float* W1_b      = (const float*)d_in[4];
  const float* W2_w      = (const float*)d_in[5];
  const float* W2_b      = (const float*)d_in[6];
  const int*   u_idx     = (const int*)d_in[7];
  const int*   i_idx     = (const int*)d_in[8];

  const int NU = in_sizes[0] / DD;   // 100000
  const int NI = in_sizes[1] / DD;   // 50000
  const int NE = in_sizes[7];        // 600000

  float* ws  = (float*)d_ws;
  float* w1u = ws;                           // [NU, 128]
  float* w1i = w1u + (size_t)NU * DD;        // [NI, 128]
  float* hu  = w1i + (size_t)NI * DD;        // [NU, 128] accumulator
  float* hi  = hu  + (size_t)NU * DD;        // [NI, 128] accumulator

  const int utiles = (NU + 15) / 16;
  const int itiles = (NI + 15) / 16;
  const int etiles = (NE + 15) / 16;

  ngcf_lin1_kernel<<<(utiles + 7) / 8, 256, 0, stream>>>(
      feat_user, W1_w, W1_b, w1u, hu, NU);
  ngcf_lin1_kernel<<<(itiles + 7) / 8, 256, 0, stream>>>(
      feat_item, W1_w, W1_b, w1i, hi, NI);

  ngcf_edge_kernel<<<(etiles + 7) / 8, 256, 0, stream>>>(
      feat_user, feat_item, edge_norm, W2_w, W2_b,
      u_idx, i_idx, w1u, w1i, hu, hi, NE);

  const int rows = NU + NI;
  ngcf_finish_kernel<<<(rows + 7) / 8, 256, 0, stream>>>(
      hu, hi, (float*)d_out, rows, NU);
}